// GravNet_73976516706975
// MI455X (gfx1250) — compile-verified
//
#include <hip/hip_runtime.h>
#include <math.h>

typedef __attribute__((ext_vector_type(2))) float v2f;
typedef __attribute__((ext_vector_type(8))) float v8f;

#define NODES   1024
#define BGRAPH  128
#define MTOT    (NODES * BGRAPH)
#define KNN     12

__device__ __forceinline__ float apply_act(float v, int act) {
    if (act == 1) return v > 0.f ? v : (expf(v) - 1.f);   // ELU (alpha=1)
    if (act == 2) return tanhf(v);
    if (act == 3) return fmaxf(v, 0.f);                   // ReLU
    return v;
}

// ---------------------------------------------------------------------------
// Weight pre-pass: transpose + zero-pad  W[Fin,Fout] -> Wt[Fout16,Fin],
// bias[Fout] -> bt[Fout16] (zeros where padded / bias==null).
// ---------------------------------------------------------------------------
__global__ void padw_kernel(const float* __restrict__ W, const float* __restrict__ bias,
                            float* __restrict__ Wt, float* __restrict__ bt,
                            int Fin, int Fout, int Fo16) {
    int i = blockIdx.x * blockDim.x + threadIdx.x;
    int total = Fo16 * Fin;
    if (i < total) {
        int n = i / Fin, k = i - n * Fin;
        Wt[i] = (n < Fout) ? W[k * Fout + n] : 0.f;
    }
    if (i < Fo16) bt[i] = (bias != nullptr && i < Fout) ? bias[i] : 0.f;
}

// ---------------------------------------------------------------------------
// Per-graph mean pool, concat [x, pooled] -> h0[B*N, 12]
// ---------------------------------------------------------------------------
__global__ __launch_bounds__(256) void pool_concat_kernel(const float* __restrict__ x,
                                                          float* __restrict__ h0) {
    __shared__ float red[256 * 6];
    __shared__ float pooled[6];
    const int b = blockIdx.x;
    const int t = threadIdx.x;
    float acc[6];
#pragma unroll
    for (int c = 0; c < 6; ++c) acc[c] = 0.f;
    for (int i = t; i < NODES; i += 256) {
        const float* xv = x + ((unsigned)(b * NODES + i)) * 6u;
#pragma unroll
        for (int c = 0; c < 6; ++c) acc[c] += xv[c];
    }
#pragma unroll
    for (int c = 0; c < 6; ++c) red[t * 6 + c] = acc[c];
    __syncthreads();
    for (int s = 128; s > 0; s >>= 1) {
        if (t < s) {
#pragma unroll
            for (int c = 0; c < 6; ++c) red[t * 6 + c] += red[(t + s) * 6 + c];
        }
        __syncthreads();
    }
    if (t < 6) pooled[t] = red[t] * (1.f / (float)NODES);
    __syncthreads();
    for (int i = t; i < NODES; i += 256) {
        const float* xv = x + ((unsigned)(b * NODES + i)) * 6u;
        float* ho = h0 + ((unsigned)(b * NODES + i)) * 12u;
#pragma unroll
        for (int c = 0; c < 6; ++c) { ho[c] = xv[c]; ho[6 + c] = pooled[c]; }
    }
}

// ---------------------------------------------------------------------------
// f32 WMMA GEMM, full tiles, no guards anywhere:
//   out[M, Sout] = act(A[M,(SA)] x Wt^T + bt),  Wt is [Fout16, FIN] transposed.
// One wave = 16x16 tile, FIN/4 chained V_WMMA_F32_16X16X4_F32.
// A 16x4: lane l -> row (l&15), K-pair ((l>>4)*2);  B 4x16 symmetric.
// C/D:    VGPR v, lane l -> row v + (l>=16 ? 8 : 0), col (l&15)
// ---------------------------------------------------------------------------
template <int FIN>
__global__ __launch_bounds__(256) void gemm_wmma_kernel(
        const float* __restrict__ A, const float* __restrict__ Wt,
        const float* __restrict__ bias, float* __restrict__ out,
        int SA, int Sout, int act, int accum) {
    const int lane = threadIdx.x & 31;
    const int wave = threadIdx.x >> 5;
    const unsigned row0 = (blockIdx.x * 8 + wave) * 16;
    const unsigned n0   = blockIdx.y * 16;
    const int col = lane & 15;
    const int hi  = lane >> 4;
    const int kp  = hi * 2;

    const float* ap = A + (size_t)(row0 + col) * (unsigned)SA + kp;
    const float* bp = Wt + (size_t)(n0 + col) * (unsigned)FIN + kp;

    const unsigned mbase = row0 + hi * 8;
    const unsigned n     = n0 + col;

    v8f c = {};
    if (accum) {
#pragma unroll
        for (int v = 0; v < 8; ++v)
            c[v] = out[(size_t)(mbase + v) * (unsigned)Sout + n];
    }
#pragma unroll
    for (int k0 = 0; k0 < FIN; k0 += 4) {
        v2f a, b;
        a.x = ap[k0]; a.y = ap[k0 + 1];
        b.x = bp[k0]; b.y = bp[k0 + 1];
        c = __builtin_amdgcn_wmma_f32_16x16x4_f32(false, a, false, b,
                                                  (short)0, c, false, false);
    }
    const float bv = bias[n];
#pragma unroll
    for (int v = 0; v < 8; ++v)
        out[(size_t)(mbase + v) * (unsigned)Sout + n] = apply_act(c[v] + bv, act);
}

// ---------------------------------------------------------------------------
// Fused pairwise-distance (WMMA Gram matrix) + top-12 + weighted mean/max agg.
// 256 threads = 8 waves; each wave owns 32 rows (one row per lane for top-k).
// s-coords staged to LDS via GLOBAL_LOAD_ASYNC_TO_LDS_B128 (ASYNCcnt path).
// The j-loop is wave-autonomous: each wave touches only its own d2s region,
// so per-wave in-order LDS semantics replace workgroup barriers.
// Grid: B * (NODES/256).  s16: [B*N,16] coords (cols 3..15 zero), hp: [B*N,16].
// ---------------------------------------------------------------------------
__global__ __launch_bounds__(256) void knn_agg_kernel(const float* __restrict__ s16,
                                                      const float* __restrict__ hp,
                                                      float* __restrict__ agg) {
    __shared__ float ssh[NODES * 4];     // graph coords, stride 4 (4th = 0)
    __shared__ float snorm[NODES];       // |s|^2
    __shared__ float d2s[256 * 17];      // d2 staging (pad 17 vs 64 banks)

    const int t    = threadIdx.x;
    const int wave = t >> 5;
    const int lane = t & 31;
    const int g     = blockIdx.x >> 2;   // graph
    const int chunk = blockIdx.x & 3;    // 256-row chunk within graph
    const unsigned gbase = (unsigned)g * NODES;

    // ---- async stage: 16B per node (3 coords + zero pad) -> LDS ----
    {
        const unsigned long long base = (unsigned long long)(s16 + (size_t)gbase * 16u);
        for (int i = t; i < NODES; i += 256) {
            unsigned lds  = (unsigned)(unsigned long long)&ssh[i * 4];
            unsigned goff = (unsigned)i * 64u;     // 16 floats * 4B per node
            asm volatile("global_load_async_to_lds_b128 %0, %1, %2"
                         :: "v"(lds), "v"(goff), "s"(base) : "memory");
        }
        asm volatile("s_wait_asynccnt 0x0" ::: "memory");
    }
    for (int i = t; i < NODES; i += 256) {
        float c0 = ssh[i * 4 + 0], c1 = ssh[i * 4 + 1], c2 = ssh[i * 4 + 2];
        snorm[i] = c0 * c0 + c1 * c1 + c2 * c2;
    }
    __syncthreads();   // ssh + snorm visible to all waves (read-only afterwards)

    const int rowl0 = chunk * 256 + wave * 32;  // wave's first row (graph-local)
    const int col = lane & 15;
    const int hi  = lane >> 4;
    const int kp  = hi * 2;

    v2f a0, a1;                                  // A tiles: rows rowl0..+15, +16..+31
    a0.x = ssh[(rowl0 + col) * 4 + kp];
    a0.y = ssh[(rowl0 + col) * 4 + kp + 1];
    a1.x = ssh[(rowl0 + 16 + col) * 4 + kp];
    a1.y = ssh[(rowl0 + 16 + col) * 4 + kp + 1];

    float bd[KNN]; int bix[KNN];
#pragma unroll
    for (int k = 0; k < KNN; ++k) { bd[k] = 3.0e38f; bix[k] = 0; }

    const int wbase = wave * 32;
    const int myrow = wbase + lane;              // LDS row owned by this lane

    for (int jt = 0; jt < NODES / 16; ++jt) {
        const int j0 = jt * 16;
        v2f bf;
        bf.x = ssh[(j0 + col) * 4 + kp];
        bf.y = ssh[(j0 + col) * 4 + kp + 1];
        v8f c0 = {}; v8f c1 = {};
        c0 = __builtin_amdgcn_wmma_f32_16x16x4_f32(false, a0, false, bf,
                                                   (short)0, c0, false, false);
        c1 = __builtin_amdgcn_wmma_f32_16x16x4_f32(false, a1, false, bf,
                                                   (short)0, c1, false, false);
        float ncol = snorm[j0 + col];
#pragma unroll
        for (int v = 0; v < 8; ++v) {
            int lr = v + hi * 8;
            d2s[(wbase + lr) * 17 + col]      = snorm[rowl0 + lr]      + ncol - 2.f * c0[v];
            d2s[(wbase + 16 + lr) * 17 + col] = snorm[rowl0 + 16 + lr] + ncol - 2.f * c1[v];
        }
        // wave-private LDS region: per-wave in-order DS ops make the data
        // visible; only prevent compiler reordering.
        __builtin_amdgcn_wave_barrier();
#pragma unroll
        for (int u = 0; u < 16; ++u) {
            float d = d2s[myrow * 17 + u];
            int j = j0 + u;
            if (d < bd[KNN - 1]) {               // early reject vs current 12th
#pragma unroll
                for (int k = 0; k < KNN; ++k) {
                    if (d < bd[k]) {
                        float td = bd[k]; int ti = bix[k];
                        bd[k] = d; bix[k] = j; d = td; j = ti;
                    }
                }
            }
        }
        __builtin_amdgcn_wave_barrier();
    }

    // weighted mean/max aggregation over the 12 nearest neighbors
    float mean[16], mx[16];
#pragma unroll
    for (int c = 0; c < 16; ++c) { mean[c] = 0.f; mx[c] = -3.0e38f; }
#pragma unroll
    for (int k = 0; k < KNN; ++k) {
        float wgt = expf(-10.f * bd[k]);
        const float* hv = hp + (size_t)(gbase + (unsigned)bix[k]) * 16u;
#pragma unroll
        for (int c = 0; c < 16; ++c) {
            float m = hv[c] * wgt;
            mean[c] += m;
            mx[c] = fmaxf(mx[c], m);
        }
    }
    const unsigned node = gbase + (unsigned)(chunk * 256 + myrow);
    float* ao = agg + (size_t)node * 32u;
#pragma unroll
    for (int c = 0; c < 16; ++c) {
        ao[c]      = mean[c] * (1.f / (float)KNN);
        ao[16 + c] = mx[c];
    }
}

// ---------------------------------------------------------------------------
// BatchNorm (training-mode stats over all B*N nodes, 16 channels)
// ---------------------------------------------------------------------------
__global__ void zero_kernel(float* p, int n) {
    int i = blockIdx.x * blockDim.x + threadIdx.x;
    if (i < n) p[i] = 0.f;
}

__global__ __launch_bounds__(256) void bn_stats_kernel(const float* __restrict__ x,
                                                       float* __restrict__ stats, int M) {
    int t = blockIdx.x * blockDim.x + threadIdx.x;
    int c = t & 15;
    int n = t >> 4;
    int stride = (gridDim.x * blockDim.x) >> 4;
    float s = 0.f, s2 = 0.f;
    for (int i = n; i < M; i += stride) {
        float v = x[(size_t)i * 16u + c];
        s += v; s2 += v * v;
    }
    atomicAdd(&stats[c], s);
    atomicAdd(&stats[16 + c], s2);
}

__global__ __launch_bounds__(256) void bn_apply_kernel(const float* __restrict__ x,
                                                       const float* __restrict__ stats,
                                                       const float* __restrict__ gam,
                                                       const float* __restrict__ bet,
                                                       float* __restrict__ outf, int M) {
    int t = blockIdx.x * blockDim.x + threadIdx.x;
    int c = t & 15;
    int n = t >> 4;
    if (n >= M) return;
    const float invM = 1.f / (float)M;
    float mean = stats[c] * invM;
    float var  = stats[16 + c] * invM - mean * mean;
    float y = gam[c] * (x[(size_t)n * 16u + c] - mean) * rsqrtf(var + 1e-5f) + bet[c];
    outf[(size_t)n * 48u + c] = y;   // channel offset baked into outf pointer
}

// ---------------------------------------------------------------------------
// 3-way softmax (logits at stride 16, cols 0..2)
// ---------------------------------------------------------------------------
__global__ __launch_bounds__(256) void softmax3_kernel(const float* __restrict__ z,
                                                       float* __restrict__ out, int M) {
    int n = blockIdx.x * blockDim.x + threadIdx.x;
    if (n >= M) return;
    const float* zp = z + (size_t)n * 16u;
    float a = zp[0], b = zp[1], c = zp[2];
    float m = fmaxf(a, fmaxf(b, c));
    float ea = expf(a - m), eb = expf(b - m), ec = expf(c - m);
    float inv = 1.f / (ea + eb + ec);
    float* op = out + (size_t)n * 3u;
    op[0] = ea * inv; op[1] = eb * inv; op[2] = ec * inv;
}

// ---------------------------------------------------------------------------
// Host-side dispatch
// ---------------------------------------------------------------------------
struct PW { const float* Wt; const float* bt; };

static void launch_gemm(int Fin, const float* A, int SA, PW w, float* out,
                        int Sout, int Fo16, int act, int accum, hipStream_t s) {
    dim3 grid(MTOT / 128, Fo16 / 16), blk(256);
    switch (Fin) {
    case 4:  gemm_wmma_kernel<4><<<grid, blk, 0, s>>>(A, w.Wt, w.bt, out, SA, Sout, act, accum); break;
    case 12: gemm_wmma_kernel<12><<<grid, blk, 0, s>>>(A, w.Wt, w.bt, out, SA, Sout, act, accum); break;
    case 16: gemm_wmma_kernel<16><<<grid, blk, 0, s>>>(A, w.Wt, w.bt, out, SA, Sout, act, accum); break;
    case 32: gemm_wmma_kernel<32><<<grid, blk, 0, s>>>(A, w.Wt, w.bt, out, SA, Sout, act, accum); break;
    case 48: gemm_wmma_kernel<48><<<grid, blk, 0, s>>>(A, w.Wt, w.bt, out, SA, Sout, act, accum); break;
    case 64: gemm_wmma_kernel<64><<<grid, blk, 0, s>>>(A, w.Wt, w.bt, out, SA, Sout, act, accum); break;
    default: break;
    }
}

extern "C" void kernel_launch(void* const* d_in, const int* in_sizes, int n_in,
                              void* d_out, int out_size, void* d_ws, size_t ws_size,
                              hipStream_t stream) {
    (void)in_sizes; (void)out_size; (void)ws_size;
    const float* IN[64];
    for (int i = 0; i < n_in && i < 64; ++i) IN[i] = (const float*)d_in[i];

    const int M = MTOT;
    const size_t Ms = (size_t)M;
    float* ws = (float*)d_ws;
    // workspace layout (units of M floats)
    float* h0    = ws;              // 12
    float* tA    = ws + 12 * Ms;    // 16
    float* tB    = ws + 28 * Ms;    // 16
    float* hft   = ws + 44 * Ms;    // 16
    float* sbuf  = ws + 60 * Ms;    // 16 (3 coords + zero pad cols)
    float* hp    = ws + 76 * Ms;    // 16
    float* aggb  = ws + 92 * Ms;    // 32
    float* gout  = ws + 124 * Ms;   // 16
    float* feats = ws + 140 * Ms;   // 48
    float* stats = ws + 188 * Ms;   // 32 floats
    float* wp    = ws + 188 * Ms + 32;  // padded weights arena
    // head reuses dead regions
    float* f1 = ws;                 // 64  (aliases h0..hft+sbuf, dead then)
    float* f2 = ws + 76 * Ms;       // 16  (aliases hp)
    float* f3 = ws + 92 * Ms;       // 16  (aliases aggb)

    dim3 blk(256);
    auto padw = [&](const float* W, const float* b, int Fin, int Fout) -> PW {
        int Fo16 = (Fout + 15) & ~15;
        float* Wt = wp; wp += (size_t)Fo16 * Fin;
        float* bt = wp; wp += Fo16;
        int total = Fo16 * Fin;
        padw_kernel<<<dim3((total + 255) / 256), blk, 0, stream>>>(W, b, Wt, bt, Fin, Fout, Fo16);
        return PW{Wt, bt};
    };

    // --- pre-pass: transpose + pad all weights ---
    PW ft1[3], ft2[3], ft3[3], lins[3], linh[3], lo1[3], lo2[3];
    for (int i = 0; i < 3; ++i) {
        const int ftw = 1 + i * 6;
        const int gnb = 19 + i * 7;
        const int Fi1 = (i == 0) ? 12 : 16;
        ft1[i]  = padw(IN[ftw + 0], IN[ftw + 1], Fi1, 16);
        ft2[i]  = padw(IN[ftw + 2], IN[ftw + 3], 16, 16);
        ft3[i]  = padw(IN[ftw + 4], IN[ftw + 5], 16, 16);
        lins[i] = padw(IN[gnb + 0], IN[gnb + 1], 16, 3);
        linh[i] = padw(IN[gnb + 2], IN[gnb + 3], 16, 16);
        lo1[i]  = padw(IN[gnb + 4], nullptr,     16, 16);
        lo2[i]  = padw(IN[gnb + 5], IN[gnb + 6], 32, 16);
    }
    PW fin1 = padw(IN[46], IN[47], 48, 64);
    PW fin2 = padw(IN[48], IN[49], 64, 4);
    PW fin3 = padw(IN[50], IN[51], 4, 3);

    // --- pipeline ---
    pool_concat_kernel<<<dim3(BGRAPH), blk, 0, stream>>>(IN[0], h0);

    for (int i = 0; i < 3; ++i) {
        const int bnb = 40 + i * 2;
        const float* Ain = (i == 0) ? h0 : (feats + (size_t)(i - 1) * 16);
        const int SA  = (i == 0) ? 12 : 48;
        const int Fi1 = (i == 0) ? 12 : 16;

        launch_gemm(Fi1, Ain, SA, ft1[i], tA,   16, 16, 1, 0, stream);  // ELU
        launch_gemm(16, tA, 16,   ft2[i], tB,   16, 16, 1, 0, stream);  // ELU
        launch_gemm(16, tB, 16,   ft3[i], hft,  16, 16, 2, 0, stream);  // tanh
        launch_gemm(16, hft, 16,  lins[i], sbuf, 16, 16, 0, 0, stream); // lin_s (padded cols)
        launch_gemm(16, hft, 16,  linh[i], hp,  16, 16, 0, 0, stream);  // lin_h
        knn_agg_kernel<<<dim3(BGRAPH * 4), blk, 0, stream>>>(sbuf, hp, aggb);
        launch_gemm(16, hft, 16,  lo1[i], gout, 16, 16, 0, 0, stream);  // lin_out1
        launch_gemm(32, aggb, 32, lo2[i], gout, 16, 16, 0, 1, stream);  // + lin_out2 (accum)
        zero_kernel<<<dim3(1), dim3(32), 0, stream>>>(stats, 32);
        bn_stats_kernel<<<dim3(256), blk, 0, stream>>>(gout, stats, M);
        bn_apply_kernel<<<dim3((M * 16) / 256), blk, 0, stream>>>(
            gout, stats, IN[bnb], IN[bnb + 1], feats + (size_t)i * 16, M);
    }

    launch_gemm(48, feats, 48, fin1, f1, 64, 64, 3, 0, stream); // ReLU
    launch_gemm(64, f1, 64,    fin2, f2, 16, 16, 3, 0, stream); // ReLU
    launch_gemm(4,  f2, 16,    fin3, f3, 16, 16, 0, 0, stream);
    softmax3_kernel<<<dim3(M / 256), blk, 0, stream>>>(f3, (float*)d_out, M);
}